// STConv_9972914061616
// MI455X (gfx1250) — compile-verified
//
#include <hip/hip_runtime.h>
#include <math.h>

// ---------------------------------------------------------------------------
// STConv for MI455X (gfx1250): all dense contractions on V_WMMA_F32_16X16X4_F32
// ---------------------------------------------------------------------------

typedef __attribute__((ext_vector_type(2))) float v2f;
typedef __attribute__((ext_vector_type(8))) float v8f;

#define NB   4
#define TIN  16
#define NN   5000
#define CIN  16
#define HID  32
#define COUT 64
#define KSZ  3
#define T1   14      // TIN - KSZ + 1
#define T2   12      // T1  - KSZ + 1
#define NE   80000
#define BT1  (NB * T1)   // 56
#define BT2  (NB * T2)   // 48
#define NTILE ((NN + 15) / 16)  // 313

__device__ __forceinline__ v8f wmma4(v2f a, v2f b, v8f c) {
  // D = A(16x4,f32) x B(4x16,f32) + C(16x16,f32)
  return __builtin_amdgcn_wmma_f32_16x16x4_f32(false, a, false, b,
                                               (short)0, c, false, false);
}

// ----------------------------- utility kernels -----------------------------

__global__ void k_zero(float* __restrict__ p, int n) {
  int i = blockIdx.x * blockDim.x + threadIdx.x;
  int st = gridDim.x * blockDim.x;
  for (; i < n; i += st) p[i] = 0.f;
}

__global__ void k_deg(float* __restrict__ deg, const int* __restrict__ ei,
                      const float* __restrict__ ew) {
  int e = blockIdx.x * blockDim.x + threadIdx.x;
  if (e < NE) atomicAdd(&deg[ei[e]], ew[e]);
}

__global__ void k_wnorm(float* __restrict__ wn, const float* __restrict__ deg,
                        const int* __restrict__ ei, const float* __restrict__ ew) {
  int e = blockIdx.x * blockDim.x + threadIdx.x;
  if (e >= NE) return;
  int r = ei[e], c = ei[NE + e];
  float dr = deg[r], dc = deg[c];
  float ir = dr > 0.f ? rsqrtf(dr) : 0.f;
  float ic = dc > 0.f ? rsqrtf(dc) : 0.f;
  wn[e] = ir * ew[e] * ic;
}

// dst[bt,row,:] += scale * wn[e] * src[bt,col,:]   (A_norm scatter, fp32 atomics)
__global__ void k_scatter(float* __restrict__ dst, const float* __restrict__ src,
                          const int* __restrict__ ei, const float* __restrict__ wn,
                          float scale) {
  long long idx = (long long)blockIdx.x * blockDim.x + threadIdx.x;
  if (idx >= (long long)NE * BT1) return;
  int e  = (int)(idx % NE);
  int bt = (int)(idx / NE);
  float w = scale * wn[e];
  int r = ei[e], c = ei[NE + e];
  const float* s = src + ((size_t)bt * NN + c) * HID;
  float*       d = dst + ((size_t)bt * NN + r) * HID;
#pragma unroll
  for (int ch = 0; ch < HID; ch++) atomicAdd(&d[ch], w * s[ch]);
}

__global__ void k_sub(float* __restrict__ a, const float* __restrict__ b, int n) {
  int i = blockIdx.x * blockDim.x + threadIdx.x;
  int st = gridDim.x * blockDim.x;
  for (; i < n; i += st) a[i] -= b[i];
}

// ------------------- temporal conv 1: (B,16,N,16)->(B,14,N,32) -------------

__global__ void __launch_bounds__(128) k_tconv1(
    const float* __restrict__ X,
    const float* __restrict__ w1, const float* __restrict__ b1,
    const float* __restrict__ w2, const float* __restrict__ b2,
    const float* __restrict__ w3, const float* __restrict__ b3,
    float* __restrict__ out) {
  __shared__ float Bw[3][48][HID];  // [set][c+16k][o] 18KB
  for (int i = threadIdx.x; i < 48 * HID; i += 128) {
    int ck = i / HID, o = i % HID;
    int c = ck & 15, k = ck >> 4;
    int widx = o * (CIN * KSZ) + c * KSZ + k;   // W:(O,C,1,K)
    Bw[0][ck][o] = w1[widx];
    Bw[1][ck][o] = w2[widx];
    Bw[2][ck][o] = w3[widx];
  }
  __syncthreads();

  const int ntiles = NB * T1 * NTILE;  // 17528
  int wave = threadIdx.x >> 5, lane = threadIdx.x & 31;
  int half = lane >> 4, lm = lane & 15;
  int tile = blockIdx.x * 4 + wave;
  if (tile >= ntiles) return;          // wave-uniform; no barriers past here

  int b   = tile / (T1 * NTILE);
  int rem = tile % (T1 * NTILE);
  int to  = rem / NTILE;
  int n0  = (rem % NTILE) * 16;
  int row = n0 + lm;

  // cache A fragments: a[ks][j] = X[b, to+k, row, c], kk = 4*ks + 2*half + j
  float a[24];
#pragma unroll
  for (int ks = 0; ks < 12; ks++)
#pragma unroll
    for (int j = 0; j < 2; j++) {
      int kk = ks * 4 + half * 2 + j;
      int k = kk >> 4, c = kk & 15;
      a[ks * 2 + j] = (row < NN)
          ? X[(((size_t)(b * TIN + to + k)) * NN + row) * CIN + c] : 0.f;
    }

  v8f zero = {0.f, 0.f, 0.f, 0.f, 0.f, 0.f, 0.f, 0.f};
  v8f acc[3][2];
#pragma unroll
  for (int s = 0; s < 3; s++)
#pragma unroll
    for (int ot = 0; ot < 2; ot++) acc[s][ot] = zero;

#pragma unroll
  for (int s = 0; s < 3; s++)
#pragma unroll
    for (int ks = 0; ks < 12; ks++) {
      v2f af; af[0] = a[2 * ks]; af[1] = a[2 * ks + 1];
#pragma unroll
      for (int ot = 0; ot < 2; ot++) {
        v2f bf;
        bf[0] = Bw[s][ks * 4 + half * 2 + 0][ot * 16 + lm];
        bf[1] = Bw[s][ks * 4 + half * 2 + 1][ot * 16 + lm];
        acc[s][ot] = wmma4(af, bf, acc[s][ot]);
      }
    }

  // epilogue: H = relu(P*sigmoid(Q)+R)
#pragma unroll
  for (int ot = 0; ot < 2; ot++) {
    int o = ot * 16 + lm;
    float bb1 = b1[o], bb2 = b2[o], bb3 = b3[o];
#pragma unroll
    for (int r = 0; r < 8; r++) {
      int n = n0 + r + half * 8;
      if (n < NN) {
        float p  = acc[0][ot][r] + bb1;
        float q  = acc[1][ot][r] + bb2;
        float rr = acc[2][ot][r] + bb3;
        float h = p * (1.f / (1.f + __expf(-q))) + rr;
        out[(((size_t)(b * T1 + to)) * NN + n) * HID + o] = fmaxf(h, 0.f);
      }
    }
  }
}

// ----- fused Chebyshev einsum: out = relu(Tx0 W0 + Tx1 W1 + Tx2 W2 + b) ----

__global__ void __launch_bounds__(128) k_cheb(
    const float* __restrict__ Tx0, const float* __restrict__ Tx1,
    const float* __restrict__ Tx2, const float* __restrict__ W,
    const float* __restrict__ bias, float* __restrict__ out) {
  __shared__ float Bw[96][HID];   // [s*32+c][d] 12KB; cheb_W flat = (s*32+c)*32+d
  for (int i = threadIdx.x; i < 96 * HID; i += 128)
    Bw[i / HID][i % HID] = W[i];
  __syncthreads();

  int wave = threadIdx.x >> 5, lane = threadIdx.x & 31;
  int half = lane >> 4, lm = lane & 15;
  int tile = blockIdx.x * 4 + wave;
  if (tile >= (BT1 * NN) / 16) return;   // 17500 exact tiles
  size_t r0 = (size_t)tile * 16;

  float a[48];
#pragma unroll
  for (int ks = 0; ks < 24; ks++)
#pragma unroll
    for (int j = 0; j < 2; j++) {
      int kk = ks * 4 + half * 2 + j;
      int s = kk >> 5, c = kk & 31;
      const float* src = (s == 0) ? Tx0 : (s == 1) ? Tx1 : Tx2;
      a[ks * 2 + j] = src[(r0 + lm) * HID + c];
    }

  v8f zero = {0.f, 0.f, 0.f, 0.f, 0.f, 0.f, 0.f, 0.f};
  v8f acc[2]; acc[0] = zero; acc[1] = zero;
#pragma unroll
  for (int ks = 0; ks < 24; ks++) {
    v2f af; af[0] = a[2 * ks]; af[1] = a[2 * ks + 1];
#pragma unroll
    for (int ot = 0; ot < 2; ot++) {
      v2f bf;
      bf[0] = Bw[ks * 4 + half * 2 + 0][ot * 16 + lm];
      bf[1] = Bw[ks * 4 + half * 2 + 1][ot * 16 + lm];
      acc[ot] = wmma4(af, bf, acc[ot]);
    }
  }

#pragma unroll
  for (int ot = 0; ot < 2; ot++) {
    int d = ot * 16 + lm;
    float bb = bias[d];
#pragma unroll
    for (int r = 0; r < 8; r++) {
      size_t rowi = r0 + r + half * 8;
      out[rowi * HID + d] = fmaxf(acc[ot][r] + bb, 0.f);
    }
  }
}

// ------------------- temporal conv 2: (B,14,N,32)->(B,12,N,64) -------------

__global__ void __launch_bounds__(128) k_tconv2(
    const float* __restrict__ Xin,
    const float* __restrict__ w1, const float* __restrict__ b1,
    const float* __restrict__ w2, const float* __restrict__ b2,
    const float* __restrict__ w3, const float* __restrict__ b3,
    float* __restrict__ out) {
  __shared__ float Bw[96][COUT];  // 24KB, restaged per weight set

  const int ntiles = NB * T2 * NTILE;  // 15024
  int wave = threadIdx.x >> 5, lane = threadIdx.x & 31;
  int half = lane >> 4, lm = lane & 15;
  int tile = blockIdx.x * 4 + wave;
  bool valid = tile < ntiles;

  int b = 0, to = 0, n0 = 0;
  float a[48];
  if (valid) {
    b = tile / (T2 * NTILE);
    int rem = tile % (T2 * NTILE);
    to = rem / NTILE;
    n0 = (rem % NTILE) * 16;
    int row = n0 + lm;
#pragma unroll
    for (int ks = 0; ks < 24; ks++)
#pragma unroll
      for (int j = 0; j < 2; j++) {
        int kk = ks * 4 + half * 2 + j;
        int k = kk >> 5, c = kk & 31;
        a[ks * 2 + j] = (row < NN)
            ? Xin[(((size_t)(b * T1 + to + k)) * NN + row) * HID + c] : 0.f;
      }
  }

  v8f zero = {0.f, 0.f, 0.f, 0.f, 0.f, 0.f, 0.f, 0.f};
  v8f acc[3][4];
#pragma unroll
  for (int s = 0; s < 3; s++)
#pragma unroll
    for (int ot = 0; ot < 4; ot++) acc[s][ot] = zero;

#pragma unroll
  for (int s = 0; s < 3; s++) {
    const float* wp = (s == 0) ? w1 : (s == 1) ? w2 : w3;
    __syncthreads();
    for (int i = threadIdx.x; i < 96 * COUT; i += 128) {
      int ck = i / COUT, o = i % COUT;
      int c = ck & 31, k = ck >> 5;
      Bw[ck][o] = wp[o * (HID * KSZ) + c * KSZ + k];
    }
    __syncthreads();
    if (valid) {
#pragma unroll
      for (int ks = 0; ks < 24; ks++) {
        v2f af; af[0] = a[2 * ks]; af[1] = a[2 * ks + 1];
#pragma unroll
        for (int ot = 0; ot < 4; ot++) {
          v2f bf;
          bf[0] = Bw[ks * 4 + half * 2 + 0][ot * 16 + lm];
          bf[1] = Bw[ks * 4 + half * 2 + 1][ot * 16 + lm];
          acc[s][ot] = wmma4(af, bf, acc[s][ot]);
        }
      }
    }
  }

  if (!valid) return;
#pragma unroll
  for (int ot = 0; ot < 4; ot++) {
    int o = ot * 16 + lm;
    float bb1 = b1[o], bb2 = b2[o], bb3 = b3[o];
#pragma unroll
    for (int r = 0; r < 8; r++) {
      int n = n0 + r + half * 8;
      if (n < NN) {
        float p  = acc[0][ot][r] + bb1;
        float q  = acc[1][ot][r] + bb2;
        float rr = acc[2][ot][r] + bb3;
        float h = p * (1.f / (1.f + __expf(-q))) + rr;
        out[(((size_t)(b * T2 + to)) * NN + n) * COUT + o] = fmaxf(h, 0.f);
      }
    }
  }
}

// ---------------- per-node BatchNorm (in place on (B,T2,N,COUT)) -----------

__global__ void __launch_bounds__(256) k_bn(float* __restrict__ T,
                                            const float* __restrict__ gamma,
                                            const float* __restrict__ beta) {
  __shared__ float s1[256], s2[256];
  int n = blockIdx.x, tid = threadIdx.x;
  float sum = 0.f, sq = 0.f;
  for (int m = tid; m < BT2 * COUT; m += 256) {
    int bt = m >> 6, c = m & 63;
    float v = T[((size_t)bt * NN + n) * COUT + c];
    sum += v; sq += v * v;
  }
  s1[tid] = sum; s2[tid] = sq;
  __syncthreads();
  for (int off = 128; off > 0; off >>= 1) {
    if (tid < off) { s1[tid] += s1[tid + off]; s2[tid] += s2[tid + off]; }
    __syncthreads();
  }
  float inv_cnt = 1.f / (float)(BT2 * COUT);
  float mean = s1[0] * inv_cnt;
  float var  = s2[0] * inv_cnt - mean * mean;  // biased, matches reference
  float sc = gamma[n] * rsqrtf(var + 1e-5f);
  float sh = beta[n] - mean * sc;
  for (int m = tid; m < BT2 * COUT; m += 256) {
    int bt = m >> 6, c = m & 63;
    size_t ix = ((size_t)bt * NN + n) * COUT + c;
    T[ix] = T[ix] * sc + sh;
  }
}

// ---------------------------------------------------------------------------

extern "C" void kernel_launch(void* const* d_in, const int* in_sizes, int n_in,
                              void* d_out, int out_size, void* d_ws, size_t ws_size,
                              hipStream_t stream) {
  (void)in_sizes; (void)n_in; (void)out_size; (void)ws_size;
  const float* X     = (const float*)d_in[0];
  const int*   ei    = (const int*)d_in[1];
  const float* ew    = (const float*)d_in[2];
  const float* t1w1  = (const float*)d_in[3];
  const float* t1b1  = (const float*)d_in[4];
  const float* t1w2  = (const float*)d_in[5];
  const float* t1b2  = (const float*)d_in[6];
  const float* t1w3  = (const float*)d_in[7];
  const float* t1b3  = (const float*)d_in[8];
  const float* chW   = (const float*)d_in[9];
  const float* chB   = (const float*)d_in[10];
  const float* t2w1  = (const float*)d_in[11];
  const float* t2b1  = (const float*)d_in[12];
  const float* t2w2  = (const float*)d_in[13];
  const float* t2b2  = (const float*)d_in[14];
  const float* t2w3  = (const float*)d_in[15];
  const float* t2b3  = (const float*)d_in[16];
  const float* gamma = (const float*)d_in[17];
  const float* beta  = (const float*)d_in[18];
  float* out = (float*)d_out;

  const size_t SZ1 = (size_t)NB * T1 * NN * HID;  // 8,960,000 floats
  float* ws  = (float*)d_ws;
  float* Tx0 = ws;                 // temporal-conv-1 output / Cheb T_0
  float* Tx1 = ws + SZ1;           // -A Tx0
  float* Tx2 = ws + 2 * SZ1;       // -2A Tx1 - Tx0
  float* Tch = ws + 3 * SZ1;       // relu(cheb) output
  float* deg = ws + 4 * SZ1;       // N floats
  float* wn  = deg + NN;           // E floats

  // --- symmetric normalization of edge weights ---
  k_zero<<<64, 256, 0, stream>>>(deg, NN);
  k_deg<<<(NE + 255) / 256, 256, 0, stream>>>(deg, ei, ew);
  k_wnorm<<<(NE + 255) / 256, 256, 0, stream>>>(wn, deg, ei, ew);

  // --- temporal conv 1 -> Tx0 ---
  {
    int ntiles = NB * T1 * NTILE;
    k_tconv1<<<(ntiles + 3) / 4, 128, 0, stream>>>(
        X, t1w1, t1b1, t1w2, t1b2, t1w3, t1b3, Tx0);
  }

  // --- Chebyshev recurrence (lambda_max=2 => lhat(z) = -A z) ---
  int sblocks = (int)(((long long)NE * BT1 + 255) / 256);
  k_zero<<<8192, 256, 0, stream>>>(Tx1, (int)SZ1);
  k_scatter<<<sblocks, 256, 0, stream>>>(Tx1, Tx0, ei, wn, -1.f);
  k_zero<<<8192, 256, 0, stream>>>(Tx2, (int)SZ1);
  k_scatter<<<sblocks, 256, 0, stream>>>(Tx2, Tx1, ei, wn, -2.f);
  k_sub<<<8192, 256, 0, stream>>>(Tx2, Tx0, (int)SZ1);

  // --- fused cheb einsum + bias + relu -> Tch ---
  k_cheb<<<(BT1 * NN / 16) / 4, 128, 0, stream>>>(Tx0, Tx1, Tx2, chW, chB, Tch);

  // --- temporal conv 2 -> d_out (same layout as final output) ---
  {
    int ntiles = NB * T2 * NTILE;
    k_tconv2<<<(ntiles + 3) / 4, 128, 0, stream>>>(
        Tch, t2w1, t2b1, t2w2, t2b2, t2w3, t2b3, out);
  }

  // --- per-node batchnorm, in place ---
  k_bn<<<NN, 256, 0, stream>>>(out, gamma, beta);
}